// Net_7825430413939
// MI455X (gfx1250) — compile-verified
//
#include <hip/hip_runtime.h>
#include <hip/hip_bf16.h>
#include <math.h>

// SGC: out = log_softmax( (A_hat^2 X) W + b ) == log_softmax( A_hat^2 (X W) + b )
// We project first (8x less propagation traffic), keep all hop state in L2.

#define FEAT 128
#define CLS  16

typedef float v2f __attribute__((ext_vector_type(2)));
typedef float v8f __attribute__((ext_vector_type(8)));

// ---------------------------------------------------------------------------
// Kernel 1: init deg=1 (self loop), zero h1/h2 accumulators.
// ---------------------------------------------------------------------------
__global__ void sgc_init_kernel(float* __restrict__ deg,
                                float* __restrict__ h1,
                                float* __restrict__ h2,
                                int N) {
    int i = blockIdx.x * blockDim.x + threadIdx.x;
    int total = N * CLS;
    if (i < total) { h1[i] = 0.0f; h2[i] = 0.0f; }
    if (i < N)     { deg[i] = 1.0f; }
}

// ---------------------------------------------------------------------------
// Kernel 2: degree accumulation over destination (row) array.
// ---------------------------------------------------------------------------
__global__ void sgc_deg_kernel(const int* __restrict__ row,
                               float* __restrict__ deg, int E) {
    int e = blockIdx.x * blockDim.x + threadIdx.x;
    if (e < E) {
        atomicAdd(deg + row[e], 1.0f);
    }
}

// ---------------------------------------------------------------------------
// Kernel 3: dinv = rsqrt(deg)   (deg >= 1 always, self-loop)
// ---------------------------------------------------------------------------
__global__ void sgc_rsqrt_kernel(float* __restrict__ deg_dinv, int N) {
    int i = blockIdx.x * blockDim.x + threadIdx.x;
    if (i < N) deg_dinv[i] = rsqrtf(deg_dinv[i]);
}

// ---------------------------------------------------------------------------
// Kernel 4: z[N,16] = x[N,128] @ W[128,16] via v_wmma_f32_16x16x4_f32.
// One 16x16 output tile per wave32. All 32 B fragments of W preloaded.
//
// Layouts (CDNA5 ISA 7.12.2):
//   A 16x4 f32 : lane<16  -> M=lane,    v0=K0, v1=K1
//                lane>=16 -> M=lane-16, v0=K2, v1=K3
//   B 4x16 f32 : VGPR v, lanes 0-15 -> row K=v,   N=lane
//                        lanes16-31 -> row K=v+2, N=lane-16
//   C/D 16x16  : VGPR v, lanes 0-15 -> M=v, lanes 16-31 -> M=v+8, N=lane&15
// ---------------------------------------------------------------------------
__global__ __launch_bounds__(128) void sgc_gemm_wmma_kernel(
    const float* __restrict__ x, const float* __restrict__ W,
    float* __restrict__ z, int N) {
    const int lane = threadIdx.x & 31;
    const int wave = threadIdx.x >> 5;          // wave32 on gfx1250
    const int tile = blockIdx.x * 4 + wave;     // 16 rows per tile
    const int half = lane >> 4;                 // 0 or 1
    const int l16  = lane & 15;
    if (tile * 16 >= N) return;                 // wave-uniform exit

    // Preload the entire 128x16 W as 32 B-fragments (64 VGPRs), reused tile-wide.
    v2f bfrag[32];
#pragma unroll
    for (int s = 0; s < 32; ++s) {
        const int k = s * 4 + half * 2;
        bfrag[s].x = W[(k + 0) * CLS + l16];
        bfrag[s].y = W[(k + 1) * CLS + l16];
    }

    v8f acc = {};
    const int m = tile * 16 + l16;
    const float* xr = x + (size_t)m * FEAT + half * 2;  // 8B-aligned
#pragma unroll
    for (int s = 0; s < 32; ++s) {
        v2f a = *(const v2f*)(xr + s * 4);
        acc = __builtin_amdgcn_wmma_f32_16x16x4_f32(
            /*neg_a=*/false, a, /*neg_b=*/false, bfrag[s],
            /*c_mod=*/(short)0, acc, /*reuse_a=*/false, /*reuse_b=*/false);
    }

#pragma unroll
    for (int v = 0; v < 8; ++v) {
        z[((size_t)tile * 16 + v + half * 8) * CLS + l16] = acc[v];
    }
}

// ---------------------------------------------------------------------------
// Kernel 5: one propagation hop on 16-dim features.
// Work items 0..E-1  : edges   (r=row[g], c=col[g])
// Work items E..E+N-1: self-loops (r=c=g-E)
// 16 lanes handle one work item (one feature channel per lane).
// dst[r,f] += dinv[r]*dinv[c]*src[c,f]   (global_atomic_add_f32; hits L2)
// ---------------------------------------------------------------------------
__global__ __launch_bounds__(256) void sgc_hop_kernel(
    const int* __restrict__ row, const int* __restrict__ col,
    const float* __restrict__ dinv,
    const float* __restrict__ src, float* __restrict__ dst,
    int E, int N) {
    const int g = (blockIdx.x * blockDim.x + threadIdx.x) >> 4;
    const int f = threadIdx.x & 15;
    if (g >= E + N) return;

    int r, c;
    if (g < E) {
        r = row[g];
        c = col[g];
        // Stream-ahead prefetch of the edge lists (global_prefetch_b8).
        if (f == 0 && g + 4096 < E) {
            __builtin_prefetch(row + g + 4096, 0, 1);
            __builtin_prefetch(col + g + 4096, 0, 1);
        }
    } else {
        r = c = g - E;
    }

    const float w = dinv[r] * dinv[c];
    const float v = src[(size_t)c * CLS + f] * w;
    atomicAdd(dst + (size_t)r * CLS + f, v);
}

// ---------------------------------------------------------------------------
// Kernel 6: out = log_softmax(h + b) row-wise over C=16. One thread per node.
// ---------------------------------------------------------------------------
__global__ void sgc_logsoftmax_kernel(const float* __restrict__ h,
                                      const float* __restrict__ bias,
                                      float* __restrict__ out, int N) {
    int i = blockIdx.x * blockDim.x + threadIdx.x;
    if (i >= N) return;

    float v[CLS];
    const float* hr = h + (size_t)i * CLS;
#pragma unroll
    for (int c = 0; c < CLS; ++c) v[c] = hr[c] + bias[c];

    float m = v[0];
#pragma unroll
    for (int c = 1; c < CLS; ++c) m = fmaxf(m, v[c]);

    float s = 0.0f;
#pragma unroll
    for (int c = 0; c < CLS; ++c) s += expf(v[c] - m);
    const float lse = m + logf(s);

    float* orow = out + (size_t)i * CLS;
#pragma unroll
    for (int c = 0; c < CLS; ++c) orow[c] = v[c] - lse;
}

// ---------------------------------------------------------------------------
// Host-side launcher
// ---------------------------------------------------------------------------
extern "C" void kernel_launch(void* const* d_in, const int* in_sizes, int n_in,
                              void* d_out, int out_size, void* d_ws, size_t ws_size,
                              hipStream_t stream) {
    const float* x     = (const float*)d_in[0];         // [N,128]
    const int*   edges = (const int*)d_in[1];           // [2,E] row-major
    const float* W     = (const float*)d_in[2];         // [128,16]
    const float* bias  = (const float*)d_in[3];         // [16]
    float*       out   = (float*)d_out;                 // [N,16]

    const int N = in_sizes[0] / FEAT;
    const int E = in_sizes[1] / 2;
    const int* row = edges;        // edge_index[0]
    const int* col = edges + E;    // edge_index[1]

    // Workspace partition (256B aligned slices)
    auto align_up = [](size_t v) { return (v + 255) & ~size_t(255); };
    char* ws = (char*)d_ws;
    size_t off = 0;
    float* dinv = (float*)(ws + off); off += align_up((size_t)N * sizeof(float));
    float* z    = (float*)(ws + off); off += align_up((size_t)N * CLS * sizeof(float));
    float* h1   = (float*)(ws + off); off += align_up((size_t)N * CLS * sizeof(float));
    float* h2   = (float*)(ws + off); off += align_up((size_t)N * CLS * sizeof(float));
    (void)ws_size;

    const int TB = 256;

    // 1) init deg=1, h1=h2=0
    {
        int total = N * CLS;
        sgc_init_kernel<<<(total + TB - 1) / TB, TB, 0, stream>>>(dinv, h1, h2, N);
    }
    // 2) degree over row
    sgc_deg_kernel<<<(E + TB - 1) / TB, TB, 0, stream>>>(row, dinv, E);
    // 3) dinv = rsqrt(deg)
    sgc_rsqrt_kernel<<<(N + TB - 1) / TB, TB, 0, stream>>>(dinv, N);
    // 4) z = x @ W  (WMMA f32 16x16x4)
    {
        int tiles  = (N + 15) / 16;
        int blocks = (tiles + 3) / 4;   // 4 waves (tiles) per 128-thread block
        sgc_gemm_wmma_kernel<<<blocks, 128, 0, stream>>>(x, W, z, N);
    }
    // 5) two hops: z -> h1 -> h2
    {
        long long groups  = (long long)E + N;
        long long threads = groups * 16;
        int blocks = (int)((threads + TB - 1) / TB);
        sgc_hop_kernel<<<blocks, TB, 0, stream>>>(row, col, dinv, z,  h1, E, N);
        sgc_hop_kernel<<<blocks, TB, 0, stream>>>(row, col, dinv, h1, h2, E, N);
    }
    // 6) bias + log_softmax -> out
    sgc_logsoftmax_kernel<<<(N + TB - 1) / TB, TB, 0, stream>>>(h2, bias, out, N);
}